// CriticNetwork_30777735643695
// MI455X (gfx1250) — compile-verified
//
#include <hip/hip_runtime.h>
#include <hip/hip_bf16.h>
#include <math.h>

typedef __attribute__((ext_vector_type(2))) float v2f;
typedef __attribute__((ext_vector_type(8))) float v8f;
typedef __attribute__((ext_vector_type(4))) unsigned int u32x4;
typedef __attribute__((ext_vector_type(8))) int i32x8;
typedef __attribute__((ext_vector_type(4))) int i32x4;

// Detect ROCm 7.2-style 5-arg TDM builtin (therock toolchain ships the TDM header
// and uses a 6-arg form instead; on that toolchain we fall back to plain loads).
#define HAVE_TDM5 0
#if defined(__has_builtin)
#if __has_builtin(__builtin_amdgcn_tensor_load_to_lds) && !__has_include(<hip/amd_detail/amd_gfx1250_TDM.h>)
#undef HAVE_TDM5
#define HAVE_TDM5 1
#endif
#endif

// ---- WMMA f32 16x16x4 wrapper (D = A*B + C, fp32 exact) ----
__device__ inline v8f wmma4(v2f a, v2f b, v8f c) {
  return __builtin_amdgcn_wmma_f32_16x16x4_f32(
      /*neg_a=*/false, a, /*neg_b=*/false, b,
      /*c_mod=*/(short)0, c, /*reuse_a=*/false, /*reuse_b=*/false);
}

// A fragment (16x4) from row-major M x K matrix, tile origin (m0,k0)
__device__ inline v2f frag_a(const float* p, int ld, int m0, int k0, int lane) {
  int m = m0 + (lane & 15);
  int k = k0 + ((lane >> 4) << 1);
  v2f r; r.x = p[m * ld + k]; r.y = p[m * ld + k + 1]; return r;
}
// B fragment (4x16) from row-major K x N matrix, tile origin (k0,n0)
__device__ inline v2f frag_b(const float* p, int ld, int k0, int n0, int lane) {
  int n = n0 + (lane & 15);
  int k = k0 + ((lane >> 4) << 1);
  v2f r; r.x = p[k * ld + n]; r.y = p[(k + 1) * ld + n]; return r;
}
// B fragment where B = W^T, W stored row-major [N][K] (typical weight matrix)
__device__ inline v2f frag_bt(const float* w, int ld, int k0, int n0, int lane) {
  int n = n0 + (lane & 15);
  int k = k0 + ((lane >> 4) << 1);
  v2f r; r.x = w[n * ld + k]; r.y = w[n * ld + k + 1]; return r;
}
// store C/D fragment to row-major with scale
__device__ inline void store_d(float* p, int ld, int m0, int n0, int lane, v8f acc, float scale) {
  int n = n0 + (lane & 15);
  int mb = m0 + ((lane >> 4) << 3);
#pragma unroll
  for (int v = 0; v < 8; ++v) p[(mb + v) * ld + n] = acc[v] * scale;
}

// Constants
#define BATCH 128
#define NN 32
#define OBS 128
#define ACTD 16
#define DD 144
#define DK 128
#define VAL_ELEMS (BATCH * NN * NN * 16)   // 2097152
#define W_ELEMS (BATCH * NN * NN)          // 131072

#if HAVE_TDM5
// TDM: async 2D tile load Global -> LDS with hardware row padding.
// Tile: 128 x 32 elements of 4 bytes; LDS rows padded 128 -> 132 DWORDs
// (pad_interval code 6 = every 128 DWORDs, pad_amount code 3 = 4 DWORDs).
__device__ inline void tdm_load_tile_128x32(const float* gsrc, float* lds_dst) {
  unsigned long long ga = (unsigned long long)(const void*)gsrc;
  unsigned int lds = (unsigned int)(unsigned long long)(void*)lds_dst;  // addr[31:0] = LDS offset
  u32x4 g0;
  g0.x = 1u;                                                    // count=1, user desc
  g0.y = lds;                                                   // lds_addr
  g0.z = (unsigned int)ga;                                      // global_addr[31:0]
  g0.w = (unsigned int)((ga >> 32) & 0x1FFFFFFu) | 0x80000000u; // global_addr[56:32] | type=2
  i32x8 g1;
  g1[0] = 0x07920000;          // data_size=4B | pad_enable | pad_interval=128dw | pad_amount=4dw
  g1[1] = (int)(128u << 16);   // tensor_dim0 = 128
  g1[2] = (int)(32u << 16);    // tensor_dim1 = 32
  g1[3] = (int)(128u << 16);   // tile_dim0 = 128
  g1[4] = 32;                  // tile_dim1 = 32 (tile_dim2 = 0)
  g1[5] = 128;                 // tensor_dim0_stride = 128
  g1[6] = 0;
  g1[7] = 0;
  i32x4 z = {0, 0, 0, 0};
  __builtin_amdgcn_tensor_load_to_lds(g0, g1, z, z, 0);
}
__device__ inline void tdm_wait0() {
#if defined(__has_builtin) && __has_builtin(__builtin_amdgcn_s_wait_tensorcnt)
  __builtin_amdgcn_s_wait_tensorcnt(0);
#else
  asm volatile("s_wait_tensorcnt 0x0" ::: "memory");
#endif
}
#endif

// ====================== Stage 1: per-batch attention + av ======================
__global__ __launch_bounds__(256) void critic_stage1(
    const float* __restrict__ states, const float* __restrict__ policies,
    const float* __restrict__ actions, const float* __restrict__ Wk,
    const float* __restrict__ Wq, const float* __restrict__ Wv1,
    const float* __restrict__ Wv2, float* __restrict__ out_weight,
    float* __restrict__ ws_w, float* __restrict__ ws_ava,
    float* __restrict__ ws_avp, float* __restrict__ ws_wava) {
  __shared__ float sm[14816];
  float* ST  = sm;            // 32 x 132 (states)
  float* ACT = sm + 4224;     // 32 x 17
  float* POL = sm + 4768;     // 32 x 17
  float* KL  = sm + 5312;     // 32 x 132
  float* QL  = sm + 9536;     // 32 x 132
  float* SL  = sm + 13760;    // 32 x 33 (scores -> weight)
  float* H1A = sm + 5312;     // 32 x 33 (reuses KL after phase 3)
  float* H1P = sm + 6368;     // 32 x 33
  float* AVA = sm + 7424;     // 32 x 132

  const int b = blockIdx.x;
  const int tid = threadIdx.x;
  const int lane = tid & 31;
  const int w = tid >> 5;

  // ---- load inputs to LDS ----
  for (int i = tid; i < NN * OBS; i += 256) {
    int m = i >> 7, d = i & 127;
    ST[m * 132 + d] = states[b * (NN * OBS) + i];
  }
  for (int i = tid; i < NN * ACTD; i += 256) {
    int m = i >> 4, d = i & 15;
    ACT[m * 17 + d] = actions[b * (NN * ACTD) + i];
    POL[m * 17 + d] = policies[b * (NN * ACTD) + i];
  }
  __syncthreads();

  // ---- phase 2: K = OA@Wk^T, Q = OA@Wq^T  ([32,128], K-dim 144) ----
  for (int t = w; t < 32; t += 8) {
    const bool isQ = (t >= 16);
    int tt = t & 15, mt = tt >> 3, nt = tt & 7;
    const float* Wm = isQ ? Wq : Wk;
    v8f acc = {};
#pragma unroll
    for (int k0 = 0; k0 < 128; k0 += 4) {      // states part
      v2f a = frag_a(ST, 132, mt * 16, k0, lane);
      v2f bb = frag_bt(Wm, DD, k0, nt * 16, lane);
      acc = wmma4(a, bb, acc);
    }
#pragma unroll
    for (int k0 = 128; k0 < DD; k0 += 4) {     // actions part
      v2f a = frag_a(ACT, 17, mt * 16, k0 - 128, lane);
      v2f bb = frag_bt(Wm, DD, k0, nt * 16, lane);
      acc = wmma4(a, bb, acc);
    }
    store_d(isQ ? QL : KL, 132, mt * 16, nt * 16, lane, acc, 1.0f);
  }
  __syncthreads();

  // ---- phase 3: S[j,i] = K[j] . Q[i] / sqrt(128) ----
  if (w < 4) {
    int mt = w >> 1, nt = w & 1;
    v8f acc = {};
#pragma unroll
    for (int k0 = 0; k0 < DK; k0 += 4) {
      v2f a = frag_a(KL, 132, mt * 16, k0, lane);
      v2f bb = frag_bt(QL, 132, k0, nt * 16, lane);  // B[d,i] = Q[i,d]
      acc = wmma4(a, bb, acc);
    }
    store_d(SL, 33, mt * 16, nt * 16, lane, acc, 0.08838834764831845f);
  }
  __syncthreads();

  // ---- softmax over i (rows of S) ----
  if (tid < 32) {
    float mx = -1e30f;
#pragma unroll
    for (int i = 0; i < 32; ++i) mx = fmaxf(mx, SL[tid * 33 + i]);
    float s = 0.f;
#pragma unroll
    for (int i = 0; i < 32; ++i) { float e = __expf(SL[tid * 33 + i] - mx); SL[tid * 33 + i] = e; s += e; }
    float inv = 1.f / s;
#pragma unroll
    for (int i = 0; i < 32; ++i) SL[tid * 33 + i] *= inv;
  }
  __syncthreads();
  for (int i = tid; i < NN * NN; i += 256) {
    float v = SL[(i >> 5) * 33 + (i & 31)];
    ws_w[b * (NN * NN) + i] = v;
    out_weight[b * (NN * NN) + i] = v;
  }

  // ---- phase 5: h1a = tanh(OA@Wv1^T), h1p = tanh(OP@Wv1^T) ([32,32]) ----
  {
    int mt = (w & 3) >> 1, nt = w & 1;
    const float* EX = (w < 4) ? ACT : POL;
    float* H = (w < 4) ? H1A : H1P;
    v8f acc = {};
#pragma unroll
    for (int k0 = 0; k0 < 128; k0 += 4) {
      v2f a = frag_a(ST, 132, mt * 16, k0, lane);
      v2f bb = frag_bt(Wv1, DD, k0, nt * 16, lane);
      acc = wmma4(a, bb, acc);
    }
#pragma unroll
    for (int k0 = 128; k0 < DD; k0 += 4) {
      v2f a = frag_a(EX, 17, mt * 16, k0 - 128, lane);
      v2f bb = frag_bt(Wv1, DD, k0, nt * 16, lane);
      acc = wmma4(a, bb, acc);
    }
    int n = nt * 16 + (lane & 15), mb = mt * 16 + ((lane >> 4) << 3);
#pragma unroll
    for (int v = 0; v < 8; ++v) H[(mb + v) * 33 + n] = tanhf(acc[v]);
  }
  __syncthreads();

  // ---- phase 6a: av_a = tanh(h1a@Wv2^T) -> LDS + ws ----
  for (int t = w; t < 16; t += 8) {
    int mt = t >> 3, nt = t & 7;
    v8f acc = {};
#pragma unroll
    for (int k0 = 0; k0 < 32; k0 += 4) {
      v2f a = frag_a(H1A, 33, mt * 16, k0, lane);
      v2f bb = frag_bt(Wv2, 32, k0, nt * 16, lane);
      acc = wmma4(a, bb, acc);
    }
    int n = nt * 16 + (lane & 15), mb = mt * 16 + ((lane >> 4) << 3);
#pragma unroll
    for (int v = 0; v < 8; ++v) {
      float tv = tanhf(acc[v]);
      AVA[(mb + v) * 132 + n] = tv;
      ws_ava[b * (NN * DK) + (mb + v) * DK + n] = tv;
    }
  }
  __syncthreads();

  // ---- phase 6b: wava = weight@av_a -> ws ; phase 6c: av_p -> ws ----
  for (int t = w; t < 32; t += 8) {
    if (t < 16) {
      int mt = t >> 3, nt = t & 7;
      v8f acc = {};
#pragma unroll
      for (int k0 = 0; k0 < 32; k0 += 4) {
        v2f a = frag_a(SL, 33, mt * 16, k0, lane);
        v2f bb = frag_b(AVA, 132, k0, nt * 16, lane);
        acc = wmma4(a, bb, acc);
      }
      int n = nt * 16 + (lane & 15), mb = mt * 16 + ((lane >> 4) << 3);
#pragma unroll
      for (int v = 0; v < 8; ++v)
        ws_wava[b * (NN * DK) + (mb + v) * DK + n] = acc[v];
    } else {
      int tt = t - 16, mt = tt >> 3, nt = tt & 7;
      v8f acc = {};
#pragma unroll
      for (int k0 = 0; k0 < 32; k0 += 4) {
        v2f a = frag_a(H1P, 33, mt * 16, k0, lane);
        v2f bb = frag_bt(Wv2, 32, k0, nt * 16, lane);
        acc = wmma4(a, bb, acc);
      }
      int n = nt * 16 + (lane & 15), mb = mt * 16 + ((lane >> 4) << 3);
#pragma unroll
      for (int v = 0; v < 8; ++v)
        ws_avp[b * (NN * DK) + (mb + v) * DK + n] = tanhf(acc[v]);
    }
  }
}

// ====================== Stage 2: per (b,y) aggregation + MLP ======================
__global__ __launch_bounds__(256) void critic_stage2(
    const float* __restrict__ states, const float* __restrict__ noise,
    const float* __restrict__ Wf1, const float* __restrict__ Wf2,
    const float* __restrict__ ws_w, const float* __restrict__ ws_ava,
    const float* __restrict__ ws_avp, const float* __restrict__ ws_wava,
    float* __restrict__ out_value) {
  __shared__ float sm[15776];
  float* NZ = sm;            // 32 x 132  noise[b,y] (z,d)
  float* WL = sm + 4224;     // 32 x 33   weight[b] (x,z)
  float* NF = sm + 5280;     // 32 x 260  node_features (x, 256)
  float* HS = sm + 13600;    // 32 x 68   hidden

  const int b = blockIdx.x >> 5;
  const int y = blockIdx.x & 31;
  const int tid = threadIdx.x;
  const int lane = tid & 31;
  const int w = tid >> 5;

  const float* nz = noise + ((size_t)(b * NN + y)) * (NN * DK);

#if HAVE_TDM5
  // Async DMA of the 16KB noise tile into LDS with hardware 128->132 row padding,
  // overlapped with the cooperative weight/states loads below.
  if (w == 0) tdm_load_tile_128x32(nz, NZ);
#else
  for (int i = tid; i < NN * DK; i += 256) { int z = i >> 7, d = i & 127; NZ[z * 132 + d] = nz[i]; }
#endif
  for (int i = tid; i < NN * NN; i += 256) { int x = i >> 5, z = i & 31; WL[x * 33 + z] = ws_w[b * (NN * NN) + i]; }
  for (int i = tid; i < NN * OBS; i += 256) { int x = i >> 7, d = i & 127; NF[x * 260 + d] = states[b * (NN * OBS) + i]; }
#if HAVE_TDM5
  if (w == 0) tdm_wait0();
#endif
  __syncthreads();

  const float* avp = ws_avp + (size_t)(b * NN + y) * DK;
  const float* ava = ws_ava + (size_t)(b * NN + y) * DK;

  // ---- T = weight @ noise[b,y]; agg = (T + wava + w[x,y]*(avp-ava))/32 ----
  for (int t = w; t < 16; t += 8) {
    int mt = t >> 3, nt = t & 7;
    v8f acc = {};
#pragma unroll
    for (int k0 = 0; k0 < 32; k0 += 4) {
      v2f a = frag_a(WL, 33, mt * 16, k0, lane);
      v2f bb = frag_b(NZ, 132, k0, nt * 16, lane);
      acc = wmma4(a, bb, acc);
    }
    int d = nt * 16 + (lane & 15), mb = mt * 16 + ((lane >> 4) << 3);
    float dav = avp[d] - ava[d];
#pragma unroll
    for (int v = 0; v < 8; ++v) {
      int x = mb + v;
      float agg = (acc[v] + ws_wava[b * (NN * DK) + x * DK + d] + WL[x * 33 + y] * dav) * (1.0f / 32.0f);
      NF[x * 260 + OBS + d] = agg;
    }
  }
  __syncthreads();

  // ---- H = leaky_relu(NF @ Wf1^T)  [32,64], K=256 ----
  {
    int mt = w >> 2, nt = w & 3;
    v8f acc = {};
#pragma unroll
    for (int k0 = 0; k0 < 256; k0 += 4) {
      v2f a = frag_a(NF, 260, mt * 16, k0, lane);
      v2f bb = frag_bt(Wf1, 256, k0, nt * 16, lane);
      acc = wmma4(a, bb, acc);
    }
    int n = nt * 16 + (lane & 15), mb = mt * 16 + ((lane >> 4) << 3);
#pragma unroll
    for (int v = 0; v < 8; ++v) {
      float h = acc[v];
      HS[(mb + v) * 68 + n] = (h > 0.f) ? h : 0.01f * h;
    }
  }
  __syncthreads();

  // ---- V = H @ Wf2^T  [32,16], K=64 ----
  if (w < 2) {
    int mt = w;
    v8f acc = {};
#pragma unroll
    for (int k0 = 0; k0 < 64; k0 += 4) {
      v2f a = frag_a(HS, 68, mt * 16, k0, lane);
      v2f bb = frag_bt(Wf2, 64, k0, 0, lane);
      acc = wmma4(a, bb, acc);
    }
    int c = lane & 15, mb = mt * 16 + ((lane >> 4) << 3);
#pragma unroll
    for (int v = 0; v < 8; ++v) {
      int x = mb + v;
      out_value[(((size_t)b * NN + x) * NN + y) * 16 + c] = acc[v];
    }
  }
}

extern "C" void kernel_launch(void* const* d_in, const int* in_sizes, int n_in,
                              void* d_out, int out_size, void* d_ws, size_t ws_size,
                              hipStream_t stream) {
  const float* states   = (const float*)d_in[0];
  const float* policies = (const float*)d_in[1];
  const float* actions  = (const float*)d_in[2];
  const float* noise    = (const float*)d_in[3];
  const float* Wk  = (const float*)d_in[4];
  const float* Wq  = (const float*)d_in[5];
  const float* Wv1 = (const float*)d_in[6];
  const float* Wv2 = (const float*)d_in[7];
  const float* Wf1 = (const float*)d_in[8];
  const float* Wf2 = (const float*)d_in[9];

  float* out        = (float*)d_out;
  float* out_value  = out;              // [B,N,N,16]
  float* out_weight = out + VAL_ELEMS;  // [B,N,N]

  float* ws      = (float*)d_ws;
  float* ws_w    = ws;                       // B*N*N
  float* ws_ava  = ws + W_ELEMS;             // B*N*DK
  float* ws_avp  = ws_ava + BATCH * NN * DK; // B*N*DK
  float* ws_wava = ws_avp + BATCH * NN * DK; // B*N*DK

  critic_stage1<<<BATCH, 256, 0, stream>>>(states, policies, actions, Wk, Wq, Wv1, Wv2,
                                           out_weight, ws_w, ws_ava, ws_avp, ws_wava);
  critic_stage2<<<BATCH * NN, 256, 0, stream>>>(states, noise, Wf1, Wf2,
                                                ws_w, ws_ava, ws_avp, ws_wava, out_value);
}